// TailGNN_26345329394247
// MI455X (gfx1250) — compile-verified
//
#include <hip/hip_runtime.h>
#include <math.h>

typedef __attribute__((ext_vector_type(2))) float v2f;
typedef __attribute__((ext_vector_type(8))) float v8f;

// ---------------------------------------------------------------- utilities

__global__ __launch_bounds__(256) void k_zero(float* __restrict__ p, size_t n) {
    size_t i = (size_t)blockIdx.x * blockDim.x + threadIdx.x;
    if (i < n) p[i] = 0.0f;
}

// cntRow[r] += 1 per edge (num_nb), deg[c] += 1 per edge (GCN col degree, self loop added later)
__global__ __launch_bounds__(256) void k_count(const long long* __restrict__ ei,
                                               float* __restrict__ cntRow,
                                               float* __restrict__ deg, int E) {
    int e = blockIdx.x * blockDim.x + threadIdx.x;
    if (e >= E) return;
    int r = (int)ei[e];
    int c = (int)ei[E + e];
    atomicAdd(cntRow + r, 1.0f);
    atomicAdd(deg + c, 1.0f);
}

// deg -> dinv = (deg+1)^-0.5   (self-loop makes deg >= 1, matching reference's where(deg>0))
__global__ __launch_bounds__(256) void k_dinv(float* __restrict__ deg, int n) {
    int i = blockIdx.x * blockDim.x + threadIdx.x;
    if (i < n) deg[i] = rsqrtf(deg[i] + 1.0f);
}

// nb_sum[row] += x[col]  (one block per edge, threads stride features)
__global__ __launch_bounds__(256) void k_scatter_add(const long long* __restrict__ ei,
                                                     const float* __restrict__ x,
                                                     float* __restrict__ nb, int E, int F) {
    int e = blockIdx.x;
    int r = (int)ei[e];
    int c = (int)ei[E + e];
    const float* src = x + (size_t)c * F;
    float* dst = nb + (size_t)r * F;
    for (int f = threadIdx.x; f < F; f += blockDim.x) atomicAdd(dst + f, src[f]);
}

__global__ __launch_bounds__(256) void k_nb_div(float* __restrict__ nb,
                                                const float* __restrict__ cnt,
                                                int Nn, int F) {
    size_t i = (size_t)blockIdx.x * blockDim.x + threadIdx.x;
    if (i >= (size_t)Nn * F) return;
    float c = cnt[i / F];
    nb[i] /= (c < 1.0f ? 1.0f : c);
}

// -------------------------------------------------------------- WMMA GEMM
// C[M x Nc] = A[M x K] @ W[Nc x K]^T   (+= if accumulate)
// Native fp32 WMMA: V_WMMA_F32_16X16X4_F32.
// wave tile: 16(M) x 64(N); 8 waves/block -> 128 x 64 C tile.
// K is a template parameter (512 / 256): compile-time trip count, immediate
// address offsets, no remainder loop.
// Inner loop is software-pipelined (register double-buffer): the 5 fragment
// loads for step k+4 are issued before the 4 WMMAs of step k, so the
// s_wait_loadcnt lands behind matrix work instead of stalling in front of it.
template <int K>
__global__ __launch_bounds__(256) void k_gemm_wmma(const float* __restrict__ A,
                                                   const float* __restrict__ W,
                                                   float* __restrict__ C,
                                                   int M, int Nc, int accumulate) {
    int wave = threadIdx.x >> 5;
    int lane = threadIdx.x & 31;
    int rowBase = blockIdx.y * 128 + wave * 16;
    if (rowBase >= M) return;                 // M % 16 == 0 -> whole-wave tiles only
    int colBase = blockIdx.x * 64;
    int ml = lane & 15;
    int koff = (lane < 16) ? 0 : 2;

    const float* Arow = A + (size_t)(rowBase + ml) * K + koff;
    const float* W0 = W + (size_t)(colBase + ml) * K + koff;
    const float* W1 = W0 + (size_t)16 * K;
    const float* W2 = W0 + (size_t)32 * K;
    const float* W3 = W0 + (size_t)48 * K;

    v8f acc0 = {0}, acc1 = {0}, acc2 = {0}, acc3 = {0};

    // prologue: fragments for k = 0
    v2f a  = *(const v2f*)(Arow);
    v2f b0 = *(const v2f*)(W0);
    v2f b1 = *(const v2f*)(W1);
    v2f b2 = *(const v2f*)(W2);
    v2f b3 = *(const v2f*)(W3);

#pragma unroll 4
    for (int k = 0; k < K - 4; k += 4) {
        // prefetch k+4 fragments (outstanding behind the WMMAs below)
        v2f an  = *(const v2f*)(Arow + k + 4);
        v2f bn0 = *(const v2f*)(W0 + k + 4);
        v2f bn1 = *(const v2f*)(W1 + k + 4);
        v2f bn2 = *(const v2f*)(W2 + k + 4);
        v2f bn3 = *(const v2f*)(W3 + k + 4);

        acc0 = __builtin_amdgcn_wmma_f32_16x16x4_f32(false, a, false, b0, (short)0, acc0, false, false);
        acc1 = __builtin_amdgcn_wmma_f32_16x16x4_f32(false, a, false, b1, (short)0, acc1, false, false);
        acc2 = __builtin_amdgcn_wmma_f32_16x16x4_f32(false, a, false, b2, (short)0, acc2, false, false);
        acc3 = __builtin_amdgcn_wmma_f32_16x16x4_f32(false, a, false, b3, (short)0, acc3, false, false);

        a = an; b0 = bn0; b1 = bn1; b2 = bn2; b3 = bn3;
    }
    // epilogue: last k-step
    acc0 = __builtin_amdgcn_wmma_f32_16x16x4_f32(false, a, false, b0, (short)0, acc0, false, false);
    acc1 = __builtin_amdgcn_wmma_f32_16x16x4_f32(false, a, false, b1, (short)0, acc1, false, false);
    acc2 = __builtin_amdgcn_wmma_f32_16x16x4_f32(false, a, false, b2, (short)0, acc2, false, false);
    acc3 = __builtin_amdgcn_wmma_f32_16x16x4_f32(false, a, false, b3, (short)0, acc3, false, false);

    int rOff = (lane < 16) ? 0 : 8;  // C/D: vgpr j -> M=j (lanes 0-15) / M=j+8 (lanes 16-31)
    v8f av[4] = {acc0, acc1, acc2, acc3};
#pragma unroll
    for (int t = 0; t < 4; ++t) {
#pragma unroll
        for (int j = 0; j < 8; ++j) {
            size_t idx = (size_t)(rowBase + rOff + j) * Nc + (colBase + t * 16 + ml);
            float v = av[t][j];
            C[idx] = accumulate ? (C[idx] + v) : v;
        }
    }
}

// --------------------------------------------------- relation (FiLM) epilogue
// out = ft + ((leaky02(gpre)+1)*m + leaky02(bpre)) - nb
__global__ __launch_bounds__(256) void k_relation_ep(const float* __restrict__ ft,
                                                     const float* __restrict__ gpre,
                                                     const float* __restrict__ bpre,
                                                     const float* __restrict__ nb,
                                                     const float* __restrict__ m,
                                                     float* __restrict__ outp,
                                                     size_t total, int F) {
    size_t i = (size_t)blockIdx.x * blockDim.x + threadIdx.x;
    if (i >= total) return;
    int f = (int)(i % F);
    float g = gpre[i];
    g = (g > 0.0f ? g : 0.2f * g) + 1.0f;
    float b = bpre[i];
    b = (b > 0.0f ? b : 0.2f * b);
    outp[i] = ft[i] + (g * m[f] + b) - nb[i];
}

// acc[col] += dinv[row]*dinv[col] * h[row]
__global__ __launch_bounds__(256) void k_edge_prop(const long long* __restrict__ ei,
                                                   const float* __restrict__ h,
                                                   const float* __restrict__ dinv,
                                                   float* __restrict__ acc, int E, int F) {
    int e = blockIdx.x;
    int r = (int)ei[e];
    int c = (int)ei[E + e];
    float w = dinv[r] * dinv[c];
    const float* src = h + (size_t)r * F;
    float* dst = acc + (size_t)c * F;
    for (int f = threadIdx.x; f < F; f += blockDim.x) atomicAdd(dst + f, w * src[f]);
}

// xo = maybe_elu( (acc + dinv^2*h (self loop) + bias + hs) / (cnt+1) )
__global__ __launch_bounds__(256) void k_combine(const float* __restrict__ acc,
                                                 const float* __restrict__ h,
                                                 const float* __restrict__ hs,
                                                 const float* __restrict__ bias,
                                                 const float* __restrict__ dinv,
                                                 const float* __restrict__ cnt,
                                                 float* __restrict__ xo,
                                                 int Nn, int F, int apply_elu) {
    size_t i = (size_t)blockIdx.x * blockDim.x + threadIdx.x;
    if (i >= (size_t)Nn * F) return;
    int row = (int)(i / F);
    int f = (int)(i % F);
    float di = dinv[row];
    float v = acc[i] + di * di * h[i] + bias[f] + hs[i];
    v /= (cnt[row] + 1.0f);
    if (apply_elu) v = (v > 0.0f) ? v : (expf(v) - 1.0f);
    xo[i] = v;
}

// log_softmax over 64 classes: one wave32 per row, 2 classes/lane, shuffle reduce
__global__ __launch_bounds__(256) void k_log_softmax64(const float* __restrict__ x2,
                                                       float* __restrict__ lsm, int Nn) {
    int wave = threadIdx.x >> 5;
    int lane = threadIdx.x & 31;
    int row = blockIdx.x * 8 + wave;
    if (row >= Nn) return;
    const float* xr = x2 + (size_t)row * 64;
    float a = xr[lane];
    float b = xr[lane + 32];
    float mx = fmaxf(a, b);
    for (int o = 16; o > 0; o >>= 1) mx = fmaxf(mx, __shfl_xor(mx, o, 32));
    float s = expf(a - mx) + expf(b - mx);
    for (int o = 16; o > 0; o >>= 1) s += __shfl_xor(s, o, 32);
    float l = logf(s);
    lsm[(size_t)row * 64 + lane] = a - mx - l;
    lsm[(size_t)row * 64 + lane + 32] = b - mx - l;
}

// ------------------------------------------------------------------- driver

extern "C" void kernel_launch(void* const* d_in, const int* in_sizes, int n_in,
                              void* d_out, int out_size, void* d_ws, size_t ws_size,
                              hipStream_t stream) {
    (void)in_sizes; (void)n_in; (void)out_size; (void)ws_size;
    const int Nn = 20000, E = 160000, F1 = 512, F2 = 256, F3 = 64;

    const float* x        = (const float*)d_in[0];
    const long long* ei   = (const long long*)d_in[1];   // int64 (2, E)
    // d_in[2] = head, fixed to 0 (tail branch) in setup
    const float* r1g1 = (const float*)d_in[3];
    const float* r1g2 = (const float*)d_in[4];
    const float* r1b1 = (const float*)d_in[5];
    const float* r1b2 = (const float*)d_in[6];
    const float* r1m  = (const float*)d_in[7];
    const float* gc1w = (const float*)d_in[8];
    const float* gc1b = (const float*)d_in[9];
    const float* r2g1 = (const float*)d_in[10];
    const float* r2g2 = (const float*)d_in[11];
    const float* r2b1 = (const float*)d_in[12];
    const float* r2b2 = (const float*)d_in[13];
    const float* r2m  = (const float*)d_in[14];
    const float* gc2w = (const float*)d_in[15];
    const float* gc2b = (const float*)d_in[16];

    float* out  = (float*)d_out;
    float* x2o  = out;                               // N x 64
    float* lsmo = out + (size_t)Nn * F3;             // N x 64
    float* out1 = out + (size_t)Nn * 2 * F3;         // N x 512
    float* out2 = out1 + (size_t)Nn * F1;            // N x 256

    float* w = (float*)d_ws;
    float* cnt  = w;  w += Nn;
    float* dinv = w;  w += Nn;
    float* nb   = w;  w += (size_t)Nn * F1;
    float* gpre = w;  w += (size_t)Nn * F1;
    float* bpre = w;  w += (size_t)Nn * F1;
    float* hbuf = w;  w += (size_t)Nn * F2;
    float* hsb  = w;  w += (size_t)Nn * F2;
    float* accb = w;  w += (size_t)Nn * F2;
    float* x1   = w;  w += (size_t)Nn * F2;

#define ZERO(p, n) k_zero<<<(unsigned)(((n) + 255) / 256), 256, 0, stream>>>((p), (size_t)(n))
#define EW(n) (unsigned)(((size_t)(n) + 255) / 256), 256, 0, stream

    // degrees / symmetric norm (shared by both layers)
    ZERO(cnt, Nn);
    ZERO(dinv, Nn);
    k_count<<<(E + 255) / 256, 256, 0, stream>>>(ei, cnt, dinv, E);
    k_dinv<<<(Nn + 255) / 256, 256, 0, stream>>>(dinv, Nn);

    // ================= layer 1 =================
    ZERO(nb, (size_t)Nn * F1);
    ZERO(accb, (size_t)Nn * F2);
    k_scatter_add<<<E, 256, 0, stream>>>(ei, x, nb, E, F1);
    k_nb_div<<<EW((size_t)Nn * F1)>>>(nb, cnt, Nn, F1);

    dim3 gRel1(F1 / 64, (Nn + 127) / 128);
    k_gemm_wmma<512><<<gRel1, 256, 0, stream>>>(x,  r1g1, gpre, Nn, F1, 0);
    k_gemm_wmma<512><<<gRel1, 256, 0, stream>>>(nb, r1g2, gpre, Nn, F1, 1);
    k_gemm_wmma<512><<<gRel1, 256, 0, stream>>>(x,  r1b1, bpre, Nn, F1, 0);
    k_gemm_wmma<512><<<gRel1, 256, 0, stream>>>(nb, r1b2, bpre, Nn, F1, 1);
    k_relation_ep<<<EW((size_t)Nn * F1)>>>(x, gpre, bpre, nb, r1m, out1, (size_t)Nn * F1, F1);

    dim3 gH1(F2 / 64, (Nn + 127) / 128);
    k_gemm_wmma<512><<<gH1, 256, 0, stream>>>(x,    gc1w, hbuf, Nn, F2, 0);
    k_gemm_wmma<512><<<gH1, 256, 0, stream>>>(out1, gc1w, hsb,  Nn, F2, 0);
    k_edge_prop<<<E, 256, 0, stream>>>(ei, hbuf, dinv, accb, E, F2);
    k_combine<<<EW((size_t)Nn * F2)>>>(accb, hbuf, hsb, gc1b, dinv, cnt, x1, Nn, F2, 1);

    // ================= layer 2 =================
    ZERO(nb, (size_t)Nn * F2);
    ZERO(accb, (size_t)Nn * F3);
    k_scatter_add<<<E, 256, 0, stream>>>(ei, x1, nb, E, F2);
    k_nb_div<<<EW((size_t)Nn * F2)>>>(nb, cnt, Nn, F2);

    dim3 gRel2(F2 / 64, (Nn + 127) / 128);
    k_gemm_wmma<256><<<gRel2, 256, 0, stream>>>(x1, r2g1, gpre, Nn, F2, 0);
    k_gemm_wmma<256><<<gRel2, 256, 0, stream>>>(nb, r2g2, gpre, Nn, F2, 1);
    k_gemm_wmma<256><<<gRel2, 256, 0, stream>>>(x1, r2b1, bpre, Nn, F2, 0);
    k_gemm_wmma<256><<<gRel2, 256, 0, stream>>>(nb, r2b2, bpre, Nn, F2, 1);
    k_relation_ep<<<EW((size_t)Nn * F2)>>>(x1, gpre, bpre, nb, r2m, out2, (size_t)Nn * F2, F2);

    dim3 gH2(F3 / 64, (Nn + 127) / 128);
    k_gemm_wmma<256><<<gH2, 256, 0, stream>>>(x1,   gc2w, hbuf, Nn, F3, 0);
    k_gemm_wmma<256><<<gH2, 256, 0, stream>>>(out2, gc2w, hsb,  Nn, F3, 0);
    k_edge_prop<<<E, 256, 0, stream>>>(ei, hbuf, dinv, accb, E, F3);
    k_combine<<<EW((size_t)Nn * F3)>>>(accb, hbuf, hsb, gc2b, dinv, cnt, x2o, Nn, F3, 0);

    k_log_softmax64<<<(Nn + 7) / 8, 256, 0, stream>>>(x2o, lsmo, Nn);

#undef ZERO
#undef EW
}